// Attention_64209761075486
// MI455X (gfx1250) — compile-verified
//
#include <hip/hip_runtime.h>
#include <hip/hip_bf16.h>

// ---------------------------------------------------------------------------
// Bahdanau additive attention, fp32, MI455X (gfx1250, wave32).
//   B=32, L1=256, L2=64, D=512
// Stage 1: c = ctx @ Wc^T, a = asp @ Wa^T     (V_WMMA_F32_16X16X4_F32, 16x64/wave)
// Stage 2: scores = V . tanh(c_i + a_j), softmax over j
//          (tanh-bound; aspect projection staged via GLOBAL_LOAD_ASYNC_TO_LDS)
// Stage 3: out = alpha @ aspect               (V_WMMA_F32_16X16X4_F32, 16x64/wave)
// ---------------------------------------------------------------------------

typedef __attribute__((ext_vector_type(2))) float v2f;
typedef __attribute__((ext_vector_type(8))) float v8f;
typedef __attribute__((ext_vector_type(4))) int   v4i;

static constexpr int BB = 32;
static constexpr int L1 = 256;
static constexpr int L2 = 64;
static constexpr int DD = 512;

#if __has_builtin(__builtin_amdgcn_global_load_async_to_lds_b128) && \
    __has_builtin(__builtin_amdgcn_s_wait_asynccnt)
#define HAVE_ASYNC_LDS 1
#else
#define HAVE_ASYNC_LDS 0
#endif

// ---- fast device math -----------------------------------------------------
static __device__ __forceinline__ float fast_tanh(float x) {
#if __has_builtin(__builtin_amdgcn_tanhf)
  return __builtin_amdgcn_tanhf(x);        // v_tanh_f32 (gfx1250 TRANS op)
#else
  return __builtin_tanhf(x);
#endif
}

static __device__ __forceinline__ float fast_exp(float x) {
#if __has_builtin(__builtin_amdgcn_exp2f)
  return __builtin_amdgcn_exp2f(x * 1.4426950408889634f);  // v_exp_f32
#else
  return __expf(x);
#endif
}

static __device__ __forceinline__ float wave_sum(float v) {
#pragma unroll
  for (int m = 16; m >= 1; m >>= 1) v += __shfl_xor(v, m, 32);
  return v;
}

static __device__ __forceinline__ float wave_max(float v) {
#pragma unroll
  for (int m = 16; m >= 1; m >>= 1) v = fmaxf(v, __shfl_xor(v, m, 32));
  return v;
}

static __device__ __forceinline__ v8f wmma_f32_k4(v2f a, v2f b, v8f c) {
  // D = A(16x4) * B(4x16) + C, fp32 end to end.
  return __builtin_amdgcn_wmma_f32_16x16x4_f32(
      /*neg_a=*/false, a, /*neg_b=*/false, b,
      /*c_mod=*/(short)0, c, /*reuse_a=*/false, /*reuse_b=*/false);
}

// ---------------------------------------------------------------------------
// Kernel 1: fused projections. y = x @ W^T.
// One wave per 16x64 output strip (4 WMMA tiles, A-fragment reused 4x).
// m-tiles: [0,512) -> context rows, [512,640) -> aspect rows.
// A fragment: lane(m=lane&15) reads X[m0+m][kb + 2*(lane>>4) + {0,1}]  (float2)
// B fragment: B[k][n] = W[n][k] -> lane(n=lane&15) reads W[n0+n][kb+2*half+{0,1}]
// ---------------------------------------------------------------------------
__global__ void __launch_bounds__(256)
proj_kernel(const float* __restrict__ ctx, const float* __restrict__ asp,
            const float* __restrict__ Wc, const float* __restrict__ Wa,
            float* __restrict__ cproj, float* __restrict__ aproj) {
  const int wid  = (blockIdx.x * blockDim.x + threadIdx.x) >> 5;  // global wave
  const int mt   = wid >> 3;   // 640 m-tiles
  const int ns   = wid & 7;    // 8 n-strips of 64
  const int lane = threadIdx.x & 31;
  const int hlf  = lane >> 4;
  const int ln   = lane & 15;

  const float* X;
  const float* W;
  float*       Y;
  int m0;
  if (mt < (BB * L1) / 16) {
    X = ctx; W = Wc; Y = cproj; m0 = mt * 16;
  } else {
    X = asp; W = Wa; Y = aproj; m0 = (mt - (BB * L1) / 16) * 16;
  }
  const int n0 = ns * 64;

  const float* arow = X + (size_t)(m0 + ln) * DD + 2 * hlf;
  const float* brow = W + (size_t)(n0 + ln) * DD + 2 * hlf;

  v8f acc0 = {}, acc1 = {}, acc2 = {}, acc3 = {};
#pragma unroll 4
  for (int kb = 0; kb < DD; kb += 4) {
    v2f av = *(const v2f*)(arow + kb);
    v2f b0 = *(const v2f*)(brow + kb);
    v2f b1 = *(const v2f*)(brow + (size_t)16 * DD + kb);
    v2f b2 = *(const v2f*)(brow + (size_t)32 * DD + kb);
    v2f b3 = *(const v2f*)(brow + (size_t)48 * DD + kb);
    acc0 = wmma_f32_k4(av, b0, acc0);
    acc1 = wmma_f32_k4(av, b1, acc1);
    acc2 = wmma_f32_k4(av, b2, acc2);
    acc3 = wmma_f32_k4(av, b3, acc3);
  }

  // C/D layout: VGPR r -> row (r + 8*hlf), col = ln (+16 per tile)
  float* yout = Y + (size_t)(m0 + 8 * hlf) * DD + n0 + ln;
#pragma unroll
  for (int r = 0; r < 8; ++r) {
    yout[(size_t)r * DD +  0] = acc0[r];
    yout[(size_t)r * DD + 16] = acc1[r];
    yout[(size_t)r * DD + 32] = acc2[r];
    yout[(size_t)r * DD + 48] = acc3[r];
  }
}

// ---------------------------------------------------------------------------
// Kernel 2: scores + softmax, fused. Block = (b, 8 consecutive i).
// aproj[b] (64x512 f32 = 128KB) staged into dynamic LDS (320KB/WGP budget,
// 2 blocks/WGP) via async cache->LDS DMA when available.
// Each wave owns one i; lanes partition d stride-32 (conflict-free ds_load_b32).
// ---------------------------------------------------------------------------
__global__ void __launch_bounds__(256)
score_softmax_kernel(const float* __restrict__ cproj,
                     const float* __restrict__ aproj,
                     const float* __restrict__ V,
                     float* __restrict__ alpha) {
  extern __shared__ float a_sh[];                 // L2 * DD floats
  const int b  = blockIdx.x >> 5;                 // L1/8 = 32 i-groups per b
  const int ig = blockIdx.x & 31;
  const int tid = threadIdx.x;

  const float* src = aproj + (size_t)b * L2 * DD;
#if HAVE_ASYNC_LDS
  {
    // GLOBAL_LOAD_ASYNC_TO_LDS_B128: cache -> LDS, tracked by ASYNCcnt.
    // Builtin prototype wants v4i pointers (16B units): AS1 src, AS3 dst.
    typedef __attribute__((address_space(1))) v4i gv4i;
    typedef __attribute__((address_space(3))) v4i lv4i;
    gv4i* gsrc = (gv4i*)src;
    lv4i* ldst = (lv4i*)a_sh;
#pragma unroll 4
    for (int t = tid; t < (L2 * DD) / 4; t += 256) {
      __builtin_amdgcn_global_load_async_to_lds_b128(gsrc + t, ldst + t,
                                                     /*offset=*/0, /*cpol=*/0);
    }
    __builtin_amdgcn_s_wait_asynccnt(0);
  }
#else
  {
    const float4* s4 = (const float4*)src;
    float4* d4 = (float4*)a_sh;
#pragma unroll 4
    for (int t = tid; t < (L2 * DD) / 4; t += 256) d4[t] = s4[t];
  }
#endif
  __syncthreads();

  const int wave = tid >> 5;
  const int lane = tid & 31;
  const int i    = ig * 8 + wave;

  // Per-lane register slices of c-row and V: d = t*32 + lane.
  const float* crow = cproj + ((size_t)b * L1 + i) * DD;
  float cv[16], vv[16];
#pragma unroll
  for (int t = 0; t < 16; ++t) {
    cv[t] = crow[t * 32 + lane];
    vv[t] = V[t * 32 + lane];
  }

  float sc[2] = {0.f, 0.f};  // scores j = lane and j = lane+32
  for (int j = 0; j < L2; ++j) {
    const float* aj = a_sh + j * DD;
    float p = 0.f;
#pragma unroll
    for (int t = 0; t < 16; ++t) {
      p = fmaf(vv[t], fast_tanh(cv[t] + aj[t * 32 + lane]), p);
    }
    const float s = wave_sum(p);        // butterfly -> broadcast to all lanes
    if ((j & 31) == lane) sc[j >> 5] = s;
  }

  // Register-resident softmax over 64 (2 values per lane).
  float mx = wave_max(fmaxf(sc[0], sc[1]));
  const float e0 = fast_exp(sc[0] - mx);
  const float e1 = fast_exp(sc[1] - mx);
  const float inv = 1.0f / wave_sum(e0 + e1);

  float* arow = alpha + ((size_t)b * L1 + i) * L2;
  arow[lane]      = e0 * inv;
  arow[lane + 32] = e1 * inv;
}

// ---------------------------------------------------------------------------
// Kernel 3: out = alpha @ aspect (batched 256x64x512 GEMM), fp32 WMMA, K=64.
// One wave per 16x64 output strip (4 accumulators, A-fragment reused 4x).
// ---------------------------------------------------------------------------
__global__ void __launch_bounds__(256)
out_gemm_kernel(const float* __restrict__ alpha, const float* __restrict__ asp,
                float* __restrict__ out) {
  const int wid  = (blockIdx.x * blockDim.x + threadIdx.x) >> 5;
  const int b    = wid >> 7;        // 16 m-tiles * 8 n-strips per batch
  const int rem  = wid & 127;
  const int mt   = rem >> 3;
  const int ns   = rem & 7;
  const int lane = threadIdx.x & 31;
  const int hlf  = lane >> 4;
  const int ln   = lane & 15;
  const int m0 = mt * 16, n0 = ns * 64;

  const float* arow  = alpha + ((size_t)b * L1 + m0 + ln) * L2 + 2 * hlf;
  const float* bbase = asp + (size_t)b * L2 * DD + n0 + ln;

  v8f acc0 = {}, acc1 = {}, acc2 = {}, acc3 = {};
#pragma unroll
  for (int kb = 0; kb < L2; kb += 4) {
    v2f av = *(const v2f*)(arow + kb);            // alpha row, contiguous in k
    const size_t r0 = (size_t)(kb + 2 * hlf) * DD;
    v2f b0, b1, b2, b3;
    b0.x = bbase[r0 +  0]; b0.y = bbase[r0 + DD +  0];
    b1.x = bbase[r0 + 16]; b1.y = bbase[r0 + DD + 16];
    b2.x = bbase[r0 + 32]; b2.y = bbase[r0 + DD + 32];
    b3.x = bbase[r0 + 48]; b3.y = bbase[r0 + DD + 48];
    acc0 = wmma_f32_k4(av, b0, acc0);
    acc1 = wmma_f32_k4(av, b1, acc1);
    acc2 = wmma_f32_k4(av, b2, acc2);
    acc3 = wmma_f32_k4(av, b3, acc3);
  }

  float* yout = out + ((size_t)b * L1 + m0 + 8 * hlf) * DD + n0 + ln;
#pragma unroll
  for (int r = 0; r < 8; ++r) {
    yout[(size_t)r * DD +  0] = acc0[r];
    yout[(size_t)r * DD + 16] = acc1[r];
    yout[(size_t)r * DD + 32] = acc2[r];
    yout[(size_t)r * DD + 48] = acc3[r];
  }
}

// ---------------------------------------------------------------------------
extern "C" void kernel_launch(void* const* d_in, const int* in_sizes, int n_in,
                              void* d_out, int out_size, void* d_ws, size_t ws_size,
                              hipStream_t stream) {
  (void)in_sizes; (void)n_in; (void)out_size; (void)ws_size;
  const float* ctx = (const float*)d_in[0];   // (32,256,512)
  const float* asp = (const float*)d_in[1];   // (32,64,512)
  const float* Wc  = (const float*)d_in[2];   // (512,512)
  const float* Wa  = (const float*)d_in[3];   // (512,512)
  const float* V   = (const float*)d_in[4];   // (512,)
  float* out = (float*)d_out;                 // (32,256,512)

  float* ws    = (float*)d_ws;
  float* cproj = ws;                                   // 8192*512 f32
  float* aproj = cproj + (size_t)BB * L1 * DD;         // 2048*512 f32
  float* alpha = aproj + (size_t)BB * L2 * DD;         // 32*256*64 f32

  // 640 m-tiles * 8 n-strips = 5120 waves, 8 waves/block
  proj_kernel<<<640, 256, 0, stream>>>(ctx, asp, Wc, Wa, cproj, aproj);

  // 32 batches * 32 i-groups = 1024 blocks; 128KB dynamic LDS each
  score_softmax_kernel<<<1024, 256, L2 * DD * sizeof(float), stream>>>(
      cproj, aproj, V, alpha);

  // 32 * 16 * 8 = 4096 waves, 8 waves/block
  out_gemm_kernel<<<512, 256, 0, stream>>>(alpha, asp, out);
}